// TinyGC2L_Net_58858231824939
// MI455X (gfx1250) — compile-verified
//
#include <hip/hip_runtime.h>

#define WAVES_PER_BLOCK 4
#define TPB (WAVES_PER_BLOCK * 32)
#define CNT_C 2048
// 4-dword pad every 64 dwords -> per-lane LDS stride 68 dwords (2-way conflicts max),
// and each lane's 64-word group is contiguous + 16B aligned -> ds_load_b128-able.
#define NPAD_WORDS (CNT_C + (CNT_C / 64) * 4)   // 2176 dwords = 8704 B per wave

struct Quat { float w, x, y, z; };

__device__ __forceinline__ Quat qmul(Quat a, Quat b) {
  Quat o;
  o.w = a.w*b.w - a.x*b.x - a.y*b.y - a.z*b.z;
  o.x = a.w*b.x + a.x*b.w + a.y*b.z - a.z*b.y;
  o.y = a.w*b.y - a.x*b.z + a.y*b.w + a.z*b.x;
  o.z = a.w*b.z + a.x*b.y - a.y*b.x + a.z*b.w;
  return o;
}

// Fast renorm: scale error only affects magnitude, never the final direction,
// so a raw v_rsq_f32 suffices for all intermediate renormalizations.
__device__ __forceinline__ Quat qnorm_fast(Quat q) {
  float n2 = q.w*q.w + q.x*q.x + q.y*q.y + q.z*q.z;
  float r  = __builtin_amdgcn_rsqf(n2);
  q.w *= r; q.x *= r; q.y *= r; q.z *= r;
  return q;
}

// Precise normalize (reference semantics) -- used once, at the very end.
__device__ __forceinline__ Quat qnorm_exact(Quat q) {
  float n2  = q.w*q.w + q.x*q.x + q.y*q.y + q.z*q.z;
  float inv = 1.0f / fmaxf(sqrtf(n2), 1e-12f);
  q.w *= inv; q.x *= inv; q.y *= inv; q.z *= inv;
  return q;
}

__device__ __forceinline__ void mlp3(const float g[3],
                                     const float* W1r, const float* b1r, float a1,
                                     const float* W2r, const float* b2r, float a2,
                                     float out[3]) {
  float h[3];
#pragma unroll
  for (int o = 0; o < 3; ++o) {
    float t = fmaf(W1r[3*o+2], g[2], fmaf(W1r[3*o+1], g[1], fmaf(W1r[3*o+0], g[0], b1r[o])));
    h[o] = (t >= 0.f) ? t : a1 * t;
  }
#pragma unroll
  for (int o = 0; o < 3; ++o) {
    float t = fmaf(W2r[3*o+2], h[2], fmaf(W2r[3*o+1], h[1], fmaf(W2r[3*o+0], h[0], b2r[o])));
    out[o] = ((t >= 0.f) ? t : a2 * t) + g[o];
  }
}

// P <- P * (1, dx, dy, dz)   (no normalization)
__device__ __forceinline__ Quat qstep(Quat P, float dx, float dy, float dz) {
  Quat n;
  n.w = P.w - P.x*dx - P.y*dy - P.z*dz;
  n.x = P.x + P.w*dx + P.y*dz - P.z*dy;
  n.y = P.y + P.w*dy - P.x*dz + P.z*dx;
  n.z = P.z + P.w*dz + P.x*dy - P.y*dx;
  return n;
}

// One 4-sample chunk: 3x float4 gyro + 1x float4 ts, iFirst..3 intervals, fast renorm.
__device__ __forceinline__ void chunk4(Quat& P, float gp[3], float& tp,
                                       float4 A, float4 Bv, float4 Cv, float4 Tv, int iFirst,
                                       const float* W1r, const float* b1r, float a1v,
                                       const float* W2r, const float* b2r, float a2v) {
  float smp[4][3] = { {A.x,  A.y,  A.z }, {A.w,  Bv.x, Bv.y},
                      {Bv.z, Bv.w, Cv.x}, {Cv.y, Cv.z, Cv.w} };
  float tcs[4] = { Tv.x, Tv.y, Tv.z, Tv.w };
#pragma unroll
  for (int i = 0; i < 4; ++i) {
    if (i < iFirst) continue;                 // compile-time peeling (iFirst is 0 or 1)
    float gc[3];
    mlp3(smp[i], W1r, b1r, a1v, W2r, b2r, a2v, gc);
    float dt4 = (tcs[i] - tp) * 0.25f;
    P = qstep(P, (gp[0]+gc[0])*dt4, (gp[1]+gc[1])*dt4, (gp[2]+gc[2])*dt4);
    gp[0]=gc[0]; gp[1]=gc[1]; gp[2]=gc[2]; tp = tcs[i];
  }
  P = qnorm_fast(P);
}

__global__ __launch_bounds__(TPB) void gyro_quat_scan_kernel(
    const float* __restrict__ ts,  const float* __restrict__ gyro,
    const float* __restrict__ q0,  const float* __restrict__ W1,
    const float* __restrict__ b1,  const float* __restrict__ a1,
    const float* __restrict__ W2,  const float* __restrict__ b2,
    const float* __restrict__ a2,  float* __restrict__ out, int B)
{
  __shared__ float ts_s[WAVES_PER_BLOCK * NPAD_WORDS];
  const int wave = threadIdx.x >> 5;
  const int lane = threadIdx.x & 31;
  const int b = blockIdx.x * WAVES_PER_BLOCK + wave;
  if (b >= B) return;

  // ---- async-stage this row's 2048 timestamps into padded LDS (CDNA5 async path) ----
  unsigned ldsBase = (unsigned)(unsigned long long)(&ts_s[wave * NPAD_WORDS]);
  unsigned rowByte = (unsigned)b * (unsigned)(CNT_C * 4);
  for (int c = lane; c < CNT_C / 4; c += 32) {
    unsigned loff = ldsBase + 16u * (unsigned)c + (((unsigned)c >> 4) << 4); // +16B pad / 16 chunks
    unsigned goff = rowByte + 16u * (unsigned)c;
    asm volatile("global_load_async_to_lds_b128 %0, %1, %2"
                 :: "v"(loff), "v"(goff), "s"(ts) : "memory");
  }

  // uniform tiny-MLP weights (overlaps with async copies in flight)
  float W1r[9], W2r[9], b1r[3], b2r[3];
#pragma unroll
  for (int i = 0; i < 9; ++i) { W1r[i] = W1[i]; W2r[i] = W2[i]; }
#pragma unroll
  for (int i = 0; i < 3; ++i) { b1r[i] = b1[i]; b2r[i] = b2[i]; }
  const float a1v = a1[0], a2v = a2[0];

  asm volatile("s_wait_asynccnt 0x0" ::: "memory");

  const float*  ts_w  = &ts_s[wave * NPAD_WORDS];
  const float4* ts_w4 = reinterpret_cast<const float4*>(ts_w);  // base is 16B aligned
  const float*  grow  = gyro + (size_t)b * (size_t)(CNT_C * 3);

  // lane-local segment: samples s .. s+64 (lane 31: s .. s+63 == 2047)
  const int s = lane * 64;
  const float4* g4 = reinterpret_cast<const float4*>(grow + 3 * s);  // 768B-aligned base
  const int tsBase4 = 17 * lane;   // padded-LDS float4 index of this lane's ts group

  // Two independent interval chains per lane (halves the serial qstep latency):
  //   A-chain: intervals with cur sample s+1 .. s+31   (chunks 0..7)
  //   B-chain: intervals with cur sample s+32 .. s+64  (chunks 8..15 + boundary)
  Quat Pa = {1.f, 0.f, 0.f, 0.f}, Pb = {1.f, 0.f, 0.f, 0.f};
  float gpa[3], tpa, gpb[3], tpb;

  // Seed A with sample s (part of chunk 0); seed B independently with sample s+31.
  {
    float grB[3] = { grow[3*(s+31)+0], grow[3*(s+31)+1], grow[3*(s+31)+2] };
    mlp3(grB, W1r, b1r, a1v, W2r, b2r, a2v, gpb);
    tpb = ts_w[68 * lane + 31];
  }

  // Fused iteration 0: A chunk 0 (seed + intervals 1..3), B chunk 8 (4 intervals).
  {
    float4 A0 = g4[0], B0 = g4[1], C0 = g4[2];
    float smp0[3] = { A0.x, A0.y, A0.z };
    mlp3(smp0, W1r, b1r, a1v, W2r, b2r, a2v, gpa);
    float4 T0 = ts_w4[tsBase4];
    tpa = T0.x;
    chunk4(Pa, gpa, tpa, A0, B0, C0, T0, /*iFirst=*/1, W1r, b1r, a1v, W2r, b2r, a2v);
    chunk4(Pb, gpb, tpb, g4[24], g4[25], g4[26], ts_w4[tsBase4 + 8], 0,
           W1r, b1r, a1v, W2r, b2r, a2v);
  }

  // Fused iterations: A chunk k and B chunk k+8 interleaved (independent chains).
  for (int k = 1; k < 8; ++k) {
    chunk4(Pa, gpa, tpa, g4[3*k],      g4[3*k+1],  g4[3*k+2],  ts_w4[tsBase4 + k],     0,
           W1r, b1r, a1v, W2r, b2r, a2v);
    chunk4(Pb, gpb, tpb, g4[3*(k+8)], g4[3*(k+8)+1], g4[3*(k+8)+2], ts_w4[tsBase4 + k + 8], 0,
           W1r, b1r, a1v, W2r, b2r, a2v);
  }

  // Boundary interval (s+63 -> s+64) for lanes 0..30; lane 31 ends at 2047.
  if (lane != 31) {
    int t = s + 64;
    float gr[3] = { grow[3*t+0], grow[3*t+1], grow[3*t+2] };
    float gc[3];
    mlp3(gr, W1r, b1r, a1v, W2r, b2r, a2v, gc);
    float tc  = ts_w[68 * (lane + 1)];                // first word of next padded group
    float dt4 = (tc - tpb) * 0.25f;
    Pb = qstep(Pb, (gpb[0]+gc[0])*dt4, (gpb[1]+gc[1])*dt4, (gpb[2]+gc[2])*dt4);
    Pb = qnorm_fast(Pb);
  }

  // Merge the two lane-local chains in order (A covers the earlier intervals).
  Quat P = qnorm_fast(qmul(Pa, Pb));

  // ---- ordered Hillis-Steele product scan across the wave32: lane31 <- P0*...*P31 ----
#pragma unroll
  for (int d = 1; d < 32; d <<= 1) {
    float pw = __shfl_up(P.w, d, 32);
    float px = __shfl_up(P.x, d, 32);
    float py = __shfl_up(P.y, d, 32);
    float pz = __shfl_up(P.z, d, 32);
    if (lane >= d) { Quat a = {pw, px, py, pz}; P = qnorm_fast(qmul(a, P)); }
  }

  Quat T;
  T.w = __shfl(P.w, 31, 32);
  T.x = __shfl(P.x, 31, 32);
  T.y = __shfl(P.y, 31, 32);
  T.z = __shfl(P.z, 31, 32);

  if (lane == 0) {
    const float* qp = q0 + (size_t)b * 4;
    Quat q = { qp[0], qp[1], qp[2], qp[3] };
    Quat r = qnorm_exact(qmul(q, T));
    float4 o4 = make_float4(r.w, r.x, r.y, r.z);
    *reinterpret_cast<float4*>(out + (size_t)b * 4) = o4;
  }
}

extern "C" void kernel_launch(void* const* d_in, const int* in_sizes, int n_in,
                              void* d_out, int out_size, void* d_ws, size_t ws_size,
                              hipStream_t stream) {
  (void)n_in; (void)out_size; (void)d_ws; (void)ws_size;
  const float* ts   = (const float*)d_in[0];
  const float* gyro = (const float*)d_in[1];
  const float* q0   = (const float*)d_in[2];
  const float* W1   = (const float*)d_in[3];
  const float* b1   = (const float*)d_in[4];
  const float* a1   = (const float*)d_in[5];
  const float* W2   = (const float*)d_in[6];
  const float* b2   = (const float*)d_in[7];
  const float* a2   = (const float*)d_in[8];
  float* out = (float*)d_out;

  const int B = in_sizes[2] / 4;   // start_quat is (B, 4)
  const int grid = (B + WAVES_PER_BLOCK - 1) / WAVES_PER_BLOCK;
  gyro_quat_scan_kernel<<<grid, TPB, 0, stream>>>(ts, gyro, q0, W1, b1, a1, W2, b2, a2, out, B);
}